// encoderNet_5162550689850
// MI455X (gfx1250) — compile-verified
//
#include <hip/hip_runtime.h>

typedef __attribute__((ext_vector_type(2))) float v2f;
typedef __attribute__((ext_vector_type(4))) float v4f;
typedef __attribute__((ext_vector_type(8))) float v8f;

#define NN      2048   // nodes (GEMM M)
#define KTOT    3072   // B*L   (GEMM K)
#define NOUT    8      // GCN_OUT (GEMM N, padded to 16 in WMMA)
#define KSPLIT  24
#define KCHUNK  (KTOT / KSPLIT)   // 128 -> 32 WMMA iters per wave
#define OUT_ELEMS (128 * 24 * 2048)

// ---------------------------------------------------------------------------
// Kernel 1: conv partials via V_WMMA_F32_16X16X4_F32.
// grid = (128 M-tiles, 6), block = 128 (4 waves); wave w of block handles
// k-chunk blockIdx.y*4 + w.  part[kchunk][node][j] gets the chunk's partial.
// Inner loop has COMPILE-TIME bounds (k0 folded into base pointers) so it is
// uniform + fully unrolled: immediate-offset load clauses + 32 WMMAs, no
// EXEC-masked loop, no pointer arithmetic in steady state.
// ---------------------------------------------------------------------------
__global__ void gemm_wmma_f32(const float* __restrict__ x,
                              const float* __restrict__ Wc,
                              float* __restrict__ part) {
  const int lane   = threadIdx.x & 31;
  const int wave   = threadIdx.x >> 5;
  const int mtile  = blockIdx.x;                 // 0..127
  const int kchunk = blockIdx.y * 4 + wave;      // 0..23
  const int mbase  = mtile * 16;
  const int k0     = kchunk * KCHUNK;

  const int  n      = lane & 15;                 // col within 16-wide tile
  const int  ncl    = n & 7;                     // always-valid Wc column
  const bool nvalid = (n < NOUT);
  const int  khalf  = (lane >> 4) * 2;           // 0 or 2: K-pair per lane half

  // Per-lane base pointers with k0/khalf folded in; loop offsets are constants.
  const float* xk = x  + (size_t)(k0 + khalf) * NN   + (mbase + n);
  const float* wk = Wc + (size_t)(k0 + khalf) * NOUT + ncl;

  v8f acc = {};
#pragma unroll
  for (int i = 0; i < KCHUNK; i += 4) {
    // A tile (16x4 f32): element (m, k0+i+khalf+j) -> xk[(i+j)*NN]
    v2f a;
    a.x = xk[(size_t)(i    ) * NN];
    a.y = xk[(size_t)(i + 1) * NN];
    // B tile (4x16 f32): unconditional load, v_cndmask zero-pad for n>=8
    float t0 = wk[(i    ) * NOUT];
    float t1 = wk[(i + 1) * NOUT];
    v2f b;
    b.x = nvalid ? t0 : 0.0f;
    b.y = nvalid ? t1 : 0.0f;
    acc = __builtin_amdgcn_wmma_f32_16x16x4_f32(
        /*neg_a=*/false, a, /*neg_b=*/false, b,
        /*c_mod=*/(short)0, acc, /*reuse_a=*/false, /*reuse_b=*/false);
  }

  // C/D layout: VGPR r -> row mbase + r (+8 for lanes 16..31), col = lane&15.
  float* dst = part + (size_t)kchunk * (NN * NOUT);
  const int mrow = mbase + (lane >> 4) * 8;
  if (nvalid) {
#pragma unroll
    for (int r = 0; r < 8; ++r)
      dst[(size_t)(mrow + r) * NOUT + n] = acc[r];
  }
}

// ---------------------------------------------------------------------------
// Kernel 2: reduce 24 partials + bias, then MLP 16->32(LeakyReLU)->1.
// combined = [conv, conv]  =>  fold W1 rows j and j+8.  One thread per node.
// ---------------------------------------------------------------------------
__global__ void reduce_mlp(const float* __restrict__ part,
                           const float* __restrict__ bc,
                           const float* __restrict__ W1,
                           const float* __restrict__ b1,
                           const float* __restrict__ W2,
                           const float* __restrict__ b2,
                           float* __restrict__ per_node) {
  const int node = blockIdx.x * blockDim.x + threadIdx.x;
  if (node >= NN) return;

  float c[NOUT];
#pragma unroll
  for (int j = 0; j < NOUT; ++j) c[j] = bc[j];

  for (int kc = 0; kc < KSPLIT; ++kc) {
    const float* p = part + ((size_t)kc * NN + node) * NOUT;
#pragma unroll
    for (int j = 0; j < NOUT; ++j) c[j] += p[j];
  }

  float out = b2[0];
#pragma unroll
  for (int o = 0; o < 32; ++o) {
    float h = b1[o];
#pragma unroll
    for (int j = 0; j < NOUT; ++j)
      h += c[j] * (W1[j * 32 + o] + W1[(j + NOUT) * 32 + o]);
    h = (h > 0.0f) ? h : 0.01f * h;          // LeakyReLU(0.01)
    out += h * W2[o];
  }
  per_node[node] = out;
}

// ---------------------------------------------------------------------------
// Kernel 3: broadcast per_node[2048] over 3072 rows with 128-bit NT stores.
// per_node (8KB) stays cache-resident; output is write-only -> non-temporal.
// ---------------------------------------------------------------------------
__global__ void broadcast_nt(const float* __restrict__ per_node,
                             v4f* __restrict__ out, int n4) {
  const int i = blockIdx.x * blockDim.x + threadIdx.x;
  if (i >= n4) return;
  const v4f* p4 = (const v4f*)per_node;
  v4f v = p4[i & (NN / 4 - 1)];
  __builtin_nontemporal_store(v, &out[i]);
}

// ---------------------------------------------------------------------------
extern "C" void kernel_launch(void* const* d_in, const int* in_sizes, int n_in,
                              void* d_out, int out_size, void* d_ws, size_t ws_size,
                              hipStream_t stream) {
  // setup_inputs order: x, edge_index, edge_attr, Wc, bc, W1, b1, W2, b2
  const float* x  = (const float*)d_in[0];
  const float* Wc = (const float*)d_in[3];
  const float* bc = (const float*)d_in[4];
  const float* W1 = (const float*)d_in[5];
  const float* b1 = (const float*)d_in[6];
  const float* W2 = (const float*)d_in[7];
  const float* b2 = (const float*)d_in[8];
  float* out = (float*)d_out;

  float* part     = (float*)d_ws;                       // [24][2048][8]
  float* per_node = part + (size_t)KSPLIT * NN * NOUT;  // [2048]

  // Phase 1: fully-written partials (no zero-init, no FP atomics -> deterministic)
  gemm_wmma_f32<<<dim3(NN / 16, KSPLIT / 4), 128, 0, stream>>>(x, Wc, part);

  // Phase 2: per-node reduce + MLP
  reduce_mlp<<<NN / 256, 256, 0, stream>>>(part, bc, W1, b1, W2, b2, per_node);

  // Phase 3: 25 MB broadcast, float4 non-temporal stores
  const int n4 = OUT_ELEMS / 4;
  broadcast_nt<<<(n4 + 255) / 256, 256, 0, stream>>>(per_node, (v4f*)out, n4);
}